// GINElayer_17583596110393
// MI455X (gfx1250) — compile-verified
//
#include <hip/hip_runtime.h>

// ---------------------------------------------------------------------------
// GINE forward for MI455X (gfx1250).
// fp32 end-to-end, all GEMMs via V_WMMA_F32_16X16X4_F32 (full-precision WMMA).
// Edge GEMM fused with gather/ReLU/scatter-add to avoid materializing the
// [1M x 128] edge-feature intermediate (saves ~1 GB of HBM traffic per layer).
// ---------------------------------------------------------------------------

typedef __attribute__((ext_vector_type(2))) float v2f;
typedef __attribute__((ext_vector_type(8))) float v8f;

#define N_NODES   65536
#define N_EDGES   1048576
#define IN_FEATS  64
#define N_HIDDEN  128
#define N_CLASSES 64
#define BN_EPS    1e-5f

__device__ __forceinline__ v8f wmma_f32_k4(v2f a, v2f b, v8f c) {
  // D = A(16x4) * B(4x16) + C(16x16), fp32 in/out.
  // args: (neg_a, A, neg_b, B, c_mod, C, reuse_a, reuse_b)
  return __builtin_amdgcn_wmma_f32_16x16x4_f32(false, a, false, b, (short)0, c,
                                               false, false);
}

// ---------------------------------------------------------------------------
__global__ void zero_kernel(float* __restrict__ p, long n) {
  long i = (long)blockIdx.x * blockDim.x + threadIdx.x;
  long stride = (long)gridDim.x * blockDim.x;
  for (; i < n; i += stride) p[i] = 0.f;
}

// ---------------------------------------------------------------------------
// Fused: e = efeat @ We^T + be ; msg = relu(h[src] + e) ; agg[dst] += msg
// One wave computes a 16-edge x EOUT strip with WMMA f32 16x16x4.
// blockDim = 128 (4 waves), 64 edges per block.
template <int EOUT>
__global__ __launch_bounds__(128) void edge_fused_kernel(
    const float* __restrict__ efeat,  // [E, 64]
    const float* __restrict__ We,     // [EOUT, 64] row-major
    const float* __restrict__ be,     // [EOUT]
    const float* __restrict__ h,      // [N, EOUT]
    const int* __restrict__ src, const int* __restrict__ dst,
    float* __restrict__ agg)          // [N, EOUT] (pre-zeroed)
{
  constexpr int NT = EOUT / 16;
  constexpr int APAD = 68;  // 16B-aligned rows, conflict-free column reads
  __shared__ float Bs[64 * EOUT];   // We^T : Bs[k*EOUT + n]
  __shared__ float bes[EOUT];
  __shared__ float As[4][16 * APAD];

  const int tid = threadIdx.x;
  for (int idx = tid; idx < 64 * EOUT; idx += 128) {
    int n = idx >> 6, k = idx & 63;
    Bs[k * EOUT + n] = We[idx];
  }
  for (int idx = tid; idx < EOUT; idx += 128) bes[idx] = be[idx];
  __syncthreads();

  const int wave = tid >> 5;
  const int lane = tid & 31;
  const int hi = lane >> 4;   // half-wave select (K rows 2..3 / M rows 8..15)
  const int lm = lane & 15;
  const int edge_base = (blockIdx.x * 4 + wave) * 16;

  // Stage A tile (16 edges x 64 feats) coalesced into LDS.
  float* as = As[wave];
  {
    const float4* s4 = (const float4*)(efeat + (size_t)edge_base * 64);
    #pragma unroll
    for (int j = 0; j < 8; ++j) {
      int idx4 = j * 32 + lane;       // 0..255 float4s
      int row = idx4 >> 4, c4 = idx4 & 15;
      float4 v = s4[idx4];
      float* d = &as[row * APAD + c4 * 4];
      d[0] = v.x; d[1] = v.y; d[2] = v.z; d[3] = v.w;
    }
  }

  v8f acc[NT];
  #pragma unroll
  for (int nt = 0; nt < NT; ++nt)
    acc[nt] = (v8f){0.f, 0.f, 0.f, 0.f, 0.f, 0.f, 0.f, 0.f};

  #pragma unroll
  for (int kk = 0; kk < 64; kk += 4) {
    v2f a;
    a.x = as[lm * APAD + kk + 2 * hi];
    a.y = as[lm * APAD + kk + 2 * hi + 1];
    #pragma unroll
    for (int nt = 0; nt < NT; ++nt) {
      v2f b;
      b.x = Bs[(kk + 2 * hi) * EOUT + nt * 16 + lm];
      b.y = Bs[(kk + 2 * hi + 1) * EOUT + nt * 16 + lm];
      acc[nt] = wmma_f32_k4(a, b, acc[nt]);
    }
  }

  // Epilogue: bias + gather h[src] + relu + atomic scatter to agg[dst].
  int sIdx[8], dIdx[8];
  #pragma unroll
  for (int i = 0; i < 8; ++i) {
    int g = edge_base + i + 8 * hi;   // C layout: VGPR i holds rows i / i+8
    sIdx[i] = src[g];
    dIdx[i] = dst[g];
  }
  #pragma unroll
  for (int nt = 0; nt < NT; ++nt) {
    int col = nt * 16 + lm;
    float bias = bes[col];
    #pragma unroll
    for (int i = 0; i < 8; ++i) {
      float v = acc[nt][i] + bias + h[(size_t)sIdx[i] * EOUT + col];
      v = fmaxf(v, 0.f);
      atomicAdd(&agg[(size_t)dIdx[i] * EOUT + col], v);
    }
  }
}

// ---------------------------------------------------------------------------
// out = act( (A [+ A2]) @ W^T + b ), A:[N,K], W:[M,K], out:[N,M]
template <int K, int M, bool HAS_A2, bool RELU>
__global__ __launch_bounds__(128) void gemm_rowtile_kernel(
    const float* __restrict__ A, const float* __restrict__ A2,
    const float* __restrict__ W, const float* __restrict__ b,
    float* __restrict__ out)
{
  constexpr int NT = M / 16;
  constexpr int APAD = K + 4;
  __shared__ float Bs[K * M];       // W^T : Bs[k*M + m]
  __shared__ float bs[M];
  __shared__ float As[4][16 * APAD];

  const int tid = threadIdx.x;
  for (int idx = tid; idx < K * M; idx += 128) {
    int m = idx / K, k = idx % K;
    Bs[k * M + m] = W[idx];
  }
  for (int idx = tid; idx < M; idx += 128) bs[idx] = b[idx];
  __syncthreads();

  const int wave = tid >> 5;
  const int lane = tid & 31;
  const int hi = lane >> 4;
  const int lm = lane & 15;
  const int row_base = (blockIdx.x * 4 + wave) * 16;

  float* as = As[wave];
  {
    constexpr int NV4 = 16 * K / 4;
    const float4* a4 = (const float4*)(A + (size_t)row_base * K);
    const float4* a24 = (const float4*)(A2 ? A2 + (size_t)row_base * K : A);
    #pragma unroll
    for (int j = 0; j < NV4 / 32; ++j) {
      int idx4 = j * 32 + lane;
      int row = idx4 / (K / 4), c4 = idx4 % (K / 4);
      float4 v = a4[idx4];
      if (HAS_A2) {
        float4 v2 = a24[idx4];
        v.x += v2.x; v.y += v2.y; v.z += v2.z; v.w += v2.w;
      }
      float* d = &as[row * APAD + c4 * 4];
      d[0] = v.x; d[1] = v.y; d[2] = v.z; d[3] = v.w;
    }
  }

  v8f acc[NT];
  #pragma unroll
  for (int nt = 0; nt < NT; ++nt)
    acc[nt] = (v8f){0.f, 0.f, 0.f, 0.f, 0.f, 0.f, 0.f, 0.f};

  #pragma unroll
  for (int kk = 0; kk < K; kk += 4) {
    v2f a;
    a.x = as[lm * APAD + kk + 2 * hi];
    a.y = as[lm * APAD + kk + 2 * hi + 1];
    #pragma unroll
    for (int nt = 0; nt < NT; ++nt) {
      v2f bb;
      bb.x = Bs[(kk + 2 * hi) * M + nt * 16 + lm];
      bb.y = Bs[(kk + 2 * hi + 1) * M + nt * 16 + lm];
      acc[nt] = wmma_f32_k4(a, bb, acc[nt]);
    }
  }

  #pragma unroll
  for (int nt = 0; nt < NT; ++nt) {
    int col = nt * 16 + lm;
    float bias = bs[col];
    #pragma unroll
    for (int i = 0; i < 8; ++i) {
      int row = row_base + i + 8 * hi;
      float v = acc[nt][i] + bias;
      if (RELU) v = fmaxf(v, 0.f);
      out[(size_t)row * M + col] = v;
    }
  }
}

// ---------------------------------------------------------------------------
// Column sum / sum-of-squares over z:[n,128] -> stats[0..127]=sum, [128..255]=sumsq
__global__ void bn_stats_kernel(const float* __restrict__ z,
                                float* __restrict__ stats, int n) {
  int col = threadIdx.x;             // blockDim = 128
  int rows_per_block = n / gridDim.x;
  int r0 = blockIdx.x * rows_per_block;
  float s = 0.f, s2 = 0.f;
  for (int r = r0; r < r0 + rows_per_block; ++r) {
    float v = z[(size_t)r * 128 + col];
    s += v; s2 += v * v;
  }
  atomicAdd(&stats[col], s);
  atomicAdd(&stats[128 + col], s2);
}

__global__ void bn_apply_relu_kernel(float* __restrict__ z,
                                     const float* __restrict__ stats,
                                     const float* __restrict__ gamma,
                                     const float* __restrict__ beta, int n) {
  size_t total = (size_t)n * 128;
  size_t stride = (size_t)gridDim.x * blockDim.x;
  const float invN = 1.0f / (float)n;
  for (size_t idx = (size_t)blockIdx.x * blockDim.x + threadIdx.x; idx < total;
       idx += stride) {
    int col = (int)(idx & 127);
    float mu = stats[col] * invN;
    float var = stats[128 + col] * invN - mu * mu;
    float inv = rsqrtf(var + BN_EPS);
    float v = (z[idx] - mu) * inv * gamma[col] + beta[col];
    z[idx] = fmaxf(v, 0.f);
  }
}

// ---------------------------------------------------------------------------
extern "C" void kernel_launch(void* const* d_in, const int* in_sizes, int n_in,
                              void* d_out, int out_size, void* d_ws,
                              size_t ws_size, hipStream_t stream) {
  const float* nfeat = (const float*)d_in[0];
  const float* efeat = (const float*)d_in[1];
  const int* src = (const int*)d_in[2];
  const int* dst = (const int*)d_in[3];
  // params flattened per JAX pytree dict-key order, per layer:
  // 0:W1 1:W2 2:We 3:b1 4:b2 5:be 6:beta 7:gamma
  auto P = [&](int layer, int slot) -> const float* {
    return (const float*)d_in[4 + layer * 8 + slot];
  };

  char* ws = (char*)d_ws;
  float* stats = (float*)ws;                                    // 256 f32
  const size_t BIG = (size_t)N_NODES * 128 * sizeof(float);     // 32 MB
  float* bufA = (float*)(ws + 4096);
  float* bufB = (float*)(ws + 4096 + BIG);
  float* bufC = (float*)(ws + 4096 + 2 * BIG);                  // z1 always

  const int EBLK = N_EDGES / 64;   // 16384 blocks, 4 waves x 16 edges
  const int NBLK = N_NODES / 64;   // 1024 blocks, 4 waves x 16 rows

  // ----- Layer 0: h_in=nfeat[ N,64], e_out=64, hid=128, out=128 (relu) -----
  zero_kernel<<<1024, 256, 0, stream>>>(bufA, (long)N_NODES * 64);
  edge_fused_kernel<64><<<EBLK, 128, 0, stream>>>(
      efeat, P(0, 2), P(0, 5), nfeat, src, dst, bufA);
  gemm_rowtile_kernel<64, 128, true, false><<<NBLK, 128, 0, stream>>>(
      nfeat, bufA, P(0, 0), P(0, 3), bufC);
  zero_kernel<<<1, 256, 0, stream>>>(stats, 256);
  bn_stats_kernel<<<256, 128, 0, stream>>>(bufC, stats, N_NODES);
  bn_apply_relu_kernel<<<2048, 256, 0, stream>>>(bufC, stats, P(0, 7), P(0, 6),
                                                 N_NODES);
  gemm_rowtile_kernel<128, 128, false, true><<<NBLK, 128, 0, stream>>>(
      bufC, nullptr, P(0, 1), P(0, 4), bufB);

  // ----- Layers 1 & 2: h_dim=128, out=128 (relu) -----
  float* hin = bufB;
  float* aggbuf = bufA;
  for (int L = 1; L <= 2; ++L) {
    zero_kernel<<<1024, 256, 0, stream>>>(aggbuf, (long)N_NODES * 128);
    edge_fused_kernel<128><<<EBLK, 128, 0, stream>>>(
        efeat, P(L, 2), P(L, 5), hin, src, dst, aggbuf);
    gemm_rowtile_kernel<128, 128, true, false><<<NBLK, 128, 0, stream>>>(
        hin, aggbuf, P(L, 0), P(L, 3), bufC);
    zero_kernel<<<1, 256, 0, stream>>>(stats, 256);
    bn_stats_kernel<<<256, 128, 0, stream>>>(bufC, stats, N_NODES);
    bn_apply_relu_kernel<<<2048, 256, 0, stream>>>(bufC, stats, P(L, 7),
                                                   P(L, 6), N_NODES);
    // h_out overwrites the (now dead) agg buffer; roles swap next layer.
    gemm_rowtile_kernel<128, 128, false, true><<<NBLK, 128, 0, stream>>>(
        bufC, nullptr, P(L, 1), P(L, 4), aggbuf);
    float* t = hin; hin = aggbuf; aggbuf = t;
  }

  // ----- Layer 3: h_dim=128, out=64, NO final relu, write d_out -----
  zero_kernel<<<1024, 256, 0, stream>>>(aggbuf, (long)N_NODES * 128);
  edge_fused_kernel<128><<<EBLK, 128, 0, stream>>>(
      efeat, P(3, 2), P(3, 5), hin, src, dst, aggbuf);
  gemm_rowtile_kernel<128, 128, true, false><<<NBLK, 128, 0, stream>>>(
      hin, aggbuf, P(3, 0), P(3, 3), bufC);
  zero_kernel<<<1, 256, 0, stream>>>(stats, 256);
  bn_stats_kernel<<<256, 128, 0, stream>>>(bufC, stats, N_NODES);
  bn_apply_relu_kernel<<<2048, 256, 0, stream>>>(bufC, stats, P(3, 7), P(3, 6),
                                                 N_NODES);
  gemm_rowtile_kernel<128, 64, false, false><<<NBLK, 128, 0, stream>>>(
      bufC, nullptr, P(3, 1), P(3, 4), (float*)d_out);
}